// Cell1AccModule_9732395893066
// MI455X (gfx1250) — compile-verified
//
#include <hip/hip_runtime.h>
#include <math.h>

// ---------------- gfx1250 async global->LDS path (guarded) ----------------
#if defined(__AMDGCN__) && __has_builtin(__builtin_amdgcn_global_load_async_to_lds_b32)
#define HAVE_ASYNC_LDS 1
#endif

#if defined(__AMDGCN__) && __has_builtin(__builtin_amdgcn_s_wait_asynccnt)
#define WAIT_ASYNC(n) __builtin_amdgcn_s_wait_asynccnt(n)
#else
#define WAIT_ASYNC(n) asm volatile("s_wait_asynccnt %0" ::"i"(n))
#endif

#ifdef HAVE_ASYNC_LDS
typedef __attribute__((address_space(1))) int* gptr_i32;
typedef __attribute__((address_space(3))) int* lptr_i32;
static __device__ __forceinline__ void async_stage_b32(const float* gp, float* lp) {
    __builtin_amdgcn_global_load_async_to_lds_b32(
        (gptr_i32)const_cast<float*>(gp),
        (lptr_i32)lp,
        0 /*offset*/, 0 /*cpol*/);
}
#endif

// ---------- float <-> order-preserving u32 key ----------
static __device__ __forceinline__ unsigned fkey(float f) {
    unsigned u = __float_as_uint(f);
    return (u & 0x80000000u) ? ~u : (u | 0x80000000u);
}
static __device__ __forceinline__ float funkey(unsigned k) {
    return __uint_as_float((k & 0x80000000u) ? (k & 0x7fffffffu) : ~k);
}

// ---------- init a u32 region to a constant ----------
__global__ void init_u32_kernel(unsigned* __restrict__ p, unsigned v, int n) {
    int i = blockIdx.x * blockDim.x + threadIdx.x;
    int stride = gridDim.x * blockDim.x;
    for (; i < n; i += stride) p[i] = v;
}

// ---------- main scatter-reduce: one thread per (element, channel) ----------
// total = N * 8.  Accumulators (~58MB) stay resident in the 192MB L2, so all
// atomics resolve at L2.  The x-stream is staged into per-lane LDS slots with
// the async copy engine (double buffered) so waves never stall on LOADcnt.
__global__ void __launch_bounds__(256)
scatter_stats_kernel(const float* __restrict__ x,
                     const int* __restrict__ m1,
                     const int* __restrict__ m2,
                     unsigned* __restrict__ c1mn,
                     unsigned* __restrict__ c1mx,
                     float*    __restrict__ c1sm,
                     unsigned* __restrict__ c2mn,
                     unsigned* __restrict__ c2mx,
                     float*    __restrict__ c2sm,
                     int total) {
    const int tid    = threadIdx.x;
    const int base   = blockIdx.x * blockDim.x + tid;
    const int stride = gridDim.x * blockDim.x;

#ifdef HAVE_ASYNC_LDS
    __shared__ float buf[2][256];
    if (base < total) async_stage_b32(&x[base], &buf[0][tid]);
    int cur = 0;
    for (int i = base; i < total; i += stride) {
        int inext = i + stride;
        float v;
        if (inext < total) {
            async_stage_b32(&x[inext], &buf[cur ^ 1][tid]);  // stage next stripe
            WAIT_ASYNC(1);                                    // current stripe done
            v = buf[cur][tid];
        } else {
            WAIT_ASYNC(0);
            v = buf[cur][tid];
        }
        cur ^= 1;
#else
    for (int i = base; i < total; i += stride) {
        if (i + stride < total) __builtin_prefetch(&x[i + stride], 0, 1);
        float v = x[i];
#endif
        int e = i >> 3;  // element index
        int c = i & 7;   // channel
        unsigned k = fkey(v);
        int l1 = m1[e] - 1;  // labels are 1-based
        int l2 = m2[e] - 1;
        unsigned* a1 = c1mn + ((size_t)l1 * 8 + c);
        unsigned* b1 = c1mx + ((size_t)l1 * 8 + c);
        float*    s1 = c1sm + ((size_t)l1 * 8 + c);
        unsigned* a2 = c2mn + ((size_t)l2 * 8 + c);
        unsigned* b2 = c2mx + ((size_t)l2 * 8 + c);
        float*    s2 = c2sm + ((size_t)l2 * 8 + c);
        atomicMin(a1, k);  // global_atomic_min_u32 (no return)
        atomicMax(b1, k);  // global_atomic_max_u32
        atomicAdd(s1, v);  // global_atomic_add_f32
        atomicMin(a2, k);
        atomicMax(b2, k);
        atomicAdd(s2, v);
    }
}

// ---------- finalize: decode keys, mean = sum/size, s = exp(-size)-0.5 ----------
// out rows are 25 wide: [min(8) | max(8) | mean(8) | s]
__global__ void finalize_stats_kernel(const unsigned* __restrict__ mn,
                                      const unsigned* __restrict__ mx,
                                      const float*    __restrict__ sm,
                                      const int*      __restrict__ sizes,
                                      float* __restrict__ out, int cnt) {
    int l = blockIdx.x * blockDim.x + threadIdx.x;
    if (l >= cnt) return;
    float sz  = (float)sizes[l];
    float inv = 1.0f / sz;
    float* o = out + (size_t)l * 25;
    const unsigned* pmn = mn + (size_t)l * 8;
    const unsigned* pmx = mx + (size_t)l * 8;
    const float*    psm = sm + (size_t)l * 8;
#pragma unroll
    for (int c = 0; c < 8; ++c) o[c]      = funkey(pmn[c]);
#pragma unroll
    for (int c = 0; c < 8; ++c) o[8 + c]  = funkey(pmx[c]);
#pragma unroll
    for (int c = 0; c < 8; ++c) o[16 + c] = psm[c] * inv;
    o[24] = expf(-sz) - 0.5f;
}

// ---------- gather cell_2 stats through the edge endpoints ----------
__global__ void gather_bounds_kernel(const float* __restrict__ c2stats,
                                     const int* __restrict__ bounds,
                                     float* __restrict__ outU,
                                     float* __restrict__ outV,
                                     int c1cnt) {
    int t = blockIdx.x * blockDim.x + threadIdx.x;
    int total = c1cnt * 25;
    if (t >= total) return;
    int e = t / 25;
    int j = t - e * 25;
    int u = bounds[2 * e]     - 1;
    int v = bounds[2 * e + 1] - 1;
    outU[t] = c2stats[(size_t)u * 25 + j];
    outV[t] = c2stats[(size_t)v * 25 + j];
}

extern "C" void kernel_launch(void* const* d_in, const int* in_sizes, int n_in,
                              void* d_out, int out_size, void* d_ws, size_t ws_size,
                              hipStream_t stream) {
    (void)n_in; (void)out_size; (void)ws_size;
    const float* x      = (const float*)d_in[0];   // [N, 8]
    const int*   m1     = (const int*)  d_in[1];   // [N] 1-based
    const int*   m2     = (const int*)  d_in[2];   // [N] 1-based
    const int*   bounds = (const int*)  d_in[3];   // [C1, 2] 1-based
    const int*   s1     = (const int*)  d_in[4];   // [C1]
    const int*   s2     = (const int*)  d_in[5];   // [C2]
    float* out = (float*)d_out;

    const int n  = in_sizes[1];   // N elements
    const int c1 = in_sizes[4];   // C1 labels
    const int c2 = in_sizes[5];   // C2 labels

    // ---- workspace carve-up (all 4-byte words) ----
    // [ mins: c1*8 | c2*8 ][ zeros: c1mx c1sm c2mx c2sm ][ c2 stats: c2*25 ]
    unsigned* c1mn = (unsigned*)d_ws;
    unsigned* c2mn = c1mn + (size_t)c1 * 8;
    unsigned* c1mx = c2mn + (size_t)c2 * 8;
    float*    c1sm = (float*)(c1mx + (size_t)c1 * 8);
    unsigned* c2mx = (unsigned*)(c1sm + (size_t)c1 * 8);
    float*    c2sm = (float*)(c2mx + (size_t)c2 * 8);
    float*    c2st = c2sm + (size_t)c2 * 8;          // [c2, 25]

    const int BS = 256;

    // ---- init accumulators ----
    int n_min  = (c1 + c2) * 8;                       // -> 0xFFFFFFFF
    int n_zero = (c1 + c2) * 16;                      // maxes + sums -> 0
    {
        int blocks = (n_min + BS * 4 - 1) / (BS * 4);
        if (blocks < 1) blocks = 1;
        init_u32_kernel<<<blocks * 4, BS, 0, stream>>>(c1mn, 0xFFFFFFFFu, n_min);
    }
    {
        int blocks = (n_zero + BS * 4 - 1) / (BS * 4);
        if (blocks < 1) blocks = 1;
        init_u32_kernel<<<blocks * 4, BS, 0, stream>>>(c1mx, 0u, n_zero);
    }

    // ---- scatter-reduce over N*8 (element, channel) pairs ----
    {
        int total  = n * 8;                           // 67,108,864 for ref sizes
        int blocks = 32768;                           // grid-stride, ~8 iters/thread
        scatter_stats_kernel<<<blocks, BS, 0, stream>>>(
            x, m1, m2, c1mn, c1mx, c1sm, c2mn, c2mx, c2sm, total);
    }

    // ---- finalize cell_1 stats straight into output section 0 ----
    {
        int blocks = (c1 + BS - 1) / BS;
        finalize_stats_kernel<<<blocks, BS, 0, stream>>>(c1mn, c1mx, c1sm, s1, out, c1);
    }

    // ---- finalize cell_2 stats into scratch ----
    {
        int blocks = (c2 + BS - 1) / BS;
        finalize_stats_kernel<<<blocks, BS, 0, stream>>>(c2mn, c2mx, c2sm, s2, c2st, c2);
    }

    // ---- gather endpoint stats into output sections 1 and 2 ----
    {
        float* outU = out + (size_t)c1 * 25;
        float* outV = out + (size_t)c1 * 50;
        int total  = c1 * 25;
        int blocks = (total + BS - 1) / BS;
        gather_bounds_kernel<<<blocks, BS, 0, stream>>>(c2st, bounds, outU, outV, c1);
    }
}